// Decoder_60232621359535
// MI455X (gfx1250) — compile-verified
//
#include <hip/hip_runtime.h>
#include <hip/hip_bf16.h>

typedef __attribute__((ext_vector_type(2))) float v2f;
typedef __attribute__((ext_vector_type(8))) float v8f;

#define CH 256
#define FIN 16
#define LDS_PITCH 260  // 260 % 64 == 4 -> 16 M-rows map to 16 distinct LDS banks

// ------------------------------------------------------------------
// degree / normalization
// ------------------------------------------------------------------
__global__ void deg_init_kernel(int* __restrict__ deg, int n) {
  int i = blockIdx.x * blockDim.x + threadIdx.x;
  if (i < n) deg[i] = 1;  // self loop contributes 1
}

__global__ void deg_count_kernel(const int* __restrict__ dst, int* __restrict__ deg, int E) {
  int i = blockIdx.x * blockDim.x + threadIdx.x;
  if (i < E) atomicAdd(&deg[dst[i]], 1);
}

__global__ void dinv_kernel(const int* __restrict__ deg, float* __restrict__ dinv, int n) {
  int i = blockIdx.x * blockDim.x + threadIdx.x;
  if (i < n) dinv[i] = rsqrtf((float)deg[i]);
}

// ------------------------------------------------------------------
// GEMM1: [N,16] x [16,256] -> [N,256] using V_WMMA_F32_16X16X4_F32.
// One wave computes a 16x16 tile (4 chained wmma over K=16).
// Block = 256 threads = 8 waves; wave w covers column tiles 2w, 2w+1.
// ------------------------------------------------------------------
__global__ void gemm16_kernel(const float* __restrict__ X, const float* __restrict__ W,
                              float* __restrict__ Y, int nrows) {
  int wave = threadIdx.x >> 5;
  int lane = threadIdx.x & 31;
  int Mloc = lane & 15;     // A row / D column-lane index
  int hi   = lane >> 4;     // lanes 16-31 hold K+2/K+3 (A) and M+8 (C/D)
  int row0 = blockIdx.x * 16;
  bool full = (row0 + 16 <= nrows);   // uniform: N % 16 == 0 -> always true

  int arow = row0 + Mloc; if (arow >= nrows) arow = nrows - 1;  // clamp OOB loads
  const float* xrow = X + (size_t)arow * FIN;

#pragma unroll
  for (int t = 0; t < 2; ++t) {
    int colTile = (wave * 2 + t) * 16;
    int col = colTile + Mloc;
    v8f c = {};
#pragma unroll
    for (int k0 = 0; k0 < FIN; k0 += 4) {
      int ka = k0 + hi * 2;
      v2f a, b;
      a.x = xrow[ka];
      a.y = xrow[ka + 1];
      b.x = W[(size_t)ka * CH + col];
      b.y = W[(size_t)(ka + 1) * CH + col];
      c = __builtin_amdgcn_wmma_f32_16x16x4_f32(false, a, false, b, (short)0, c, false, false);
    }
    float* yp = Y + (size_t)(row0 + hi * 8) * CH + col;
    if (full) {
#pragma unroll
      for (int r = 0; r < 8; ++r) yp[(size_t)r * CH] = c[r];
    } else {
#pragma unroll
      for (int r = 0; r < 8; ++r) {
        int row = row0 + hi * 8 + r;
        if (row < nrows) Y[(size_t)row * CH + col] = c[r];
      }
    }
  }
}

// ------------------------------------------------------------------
// GEMM2: [N,256] x [256,256] -> [N,256]. A 16x256 panel staged in LDS
// (padded pitch), shared by the 8 waves of the block. 64 chained WMMAs
// per 16x16 output tile.
// ------------------------------------------------------------------
__global__ void gemm256_kernel(const float* __restrict__ X, const float* __restrict__ W,
                               float* __restrict__ Y, int nrows) {
  __shared__ float As[16 * LDS_PITCH];
  int tid  = threadIdx.x;
  int row0 = blockIdx.x * 16;
  bool full = (row0 + 16 <= nrows);

  for (int i = tid; i < 16 * CH; i += 256) {
    int r = i >> 8, cc = i & 255;
    int row = row0 + r; if (row >= nrows) row = nrows - 1;
    As[r * LDS_PITCH + cc] = X[(size_t)row * CH + cc];
  }
  __syncthreads();

  int wave = tid >> 5;
  int lane = tid & 31;
  int Mloc = lane & 15;
  int hi   = lane >> 4;

#pragma unroll
  for (int t = 0; t < 2; ++t) {
    int colTile = (wave * 2 + t) * 16;
    int col = colTile + Mloc;
    v8f c = {};
#pragma unroll 4
    for (int k0 = 0; k0 < CH; k0 += 4) {
      int ka = k0 + hi * 2;
      v2f a, b;
      a.x = As[Mloc * LDS_PITCH + ka];
      a.y = As[Mloc * LDS_PITCH + ka + 1];
      b.x = W[(size_t)ka * CH + col];
      b.y = W[(size_t)(ka + 1) * CH + col];
      c = __builtin_amdgcn_wmma_f32_16x16x4_f32(false, a, false, b, (short)0, c, false, false);
    }
    float* yp = Y + (size_t)(row0 + hi * 8) * CH + col;
    if (full) {
#pragma unroll
      for (int r = 0; r < 8; ++r) yp[(size_t)r * CH] = c[r];
    } else {
#pragma unroll
      for (int r = 0; r < 8; ++r) {
        int row = row0 + hi * 8 + r;
        if (row < nrows) Y[(size_t)row * CH + col] = c[r];
      }
    }
  }
}

// ------------------------------------------------------------------
// agg[n,c] = bias[c] + dinv[n]^2 * h[n,c]   (self-loop term + bias)
// float4 path: i indexes float4 quads (total4 = N*64)
// ------------------------------------------------------------------
__global__ void self_bias_kernel(const float4* __restrict__ h4, const float* __restrict__ dinv,
                                 const float4* __restrict__ bias4, float4* __restrict__ agg4,
                                 int total4) {
  int i = blockIdx.x * blockDim.x + threadIdx.x;
  if (i < total4) {
    int node = i >> 6;       // 64 float4 per node row
    int c4 = i & 63;
    float di = dinv[node];
    float n2 = di * di;
    float4 hv = h4[i];
    float4 bv = bias4[c4];
    float4 o;
    o.x = bv.x + n2 * hv.x;
    o.y = bv.y + n2 * hv.y;
    o.z = bv.z + n2 * hv.z;
    o.w = bv.w + n2 * hv.w;
    agg4[i] = o;
  }
}

// ------------------------------------------------------------------
// Edge scatter: agg[dst] += dinv[src]*dinv[dst] * h[src].
// One wave32 per edge; 2x float4 (b128) loads per lane; HW f32 atomics.
// ------------------------------------------------------------------
__global__ void scatter_kernel(const float* __restrict__ h, const int* __restrict__ src,
                               const int* __restrict__ dst, const float* __restrict__ dinv,
                               float* __restrict__ agg, int E) {
  int wave = threadIdx.x >> 5;
  int lane = threadIdx.x & 31;
  int e = blockIdx.x * 8 + wave;
  if (e >= E) return;
  int s = src[e], d = dst[e];
  float nrm = dinv[s] * dinv[d];
  const float4* hs4 = (const float4*)(h + (size_t)s * CH);
  float* ad = agg + (size_t)d * CH;
#pragma unroll
  for (int i = 0; i < 2; ++i) {
    int c4 = lane + i * 32;          // float4 index within the 256-ch row
    float4 v = hs4[c4];
    int c = c4 * 4;
    __hip_atomic_fetch_add(&ad[c + 0], nrm * v.x, __ATOMIC_RELAXED, __HIP_MEMORY_SCOPE_AGENT);
    __hip_atomic_fetch_add(&ad[c + 1], nrm * v.y, __ATOMIC_RELAXED, __HIP_MEMORY_SCOPE_AGENT);
    __hip_atomic_fetch_add(&ad[c + 2], nrm * v.z, __ATOMIC_RELAXED, __HIP_MEMORY_SCOPE_AGENT);
    __hip_atomic_fetch_add(&ad[c + 3], nrm * v.w, __ATOMIC_RELAXED, __HIP_MEMORY_SCOPE_AGENT);
  }
}

// ------------------------------------------------------------------
// Decoder: out[e] = sigmoid(dot(h[src], h[dst])). float4 loads, wave32 reduce.
// ------------------------------------------------------------------
__global__ void edge_dot_kernel(const float* __restrict__ h, const int* __restrict__ src,
                                const int* __restrict__ dst, float* __restrict__ out, int E) {
  int wave = threadIdx.x >> 5;
  int lane = threadIdx.x & 31;
  int e = blockIdx.x * 8 + wave;
  if (e >= E) return;
  int s = src[e], d = dst[e];
  const float4* hs4 = (const float4*)(h + (size_t)s * CH);
  const float4* hd4 = (const float4*)(h + (size_t)d * CH);
  float acc = 0.f;
#pragma unroll
  for (int i = 0; i < 2; ++i) {
    int c4 = lane + i * 32;
    float4 a = hs4[c4];
    float4 b = hd4[c4];
    acc += a.x * b.x + a.y * b.y + a.z * b.z + a.w * b.w;
  }
#pragma unroll
  for (int off = 16; off > 0; off >>= 1) acc += __shfl_xor(acc, off, 32);
  if (lane == 0) out[e] = 1.0f / (1.0f + __expf(-acc));
}

// ------------------------------------------------------------------
extern "C" void kernel_launch(void* const* d_in, const int* in_sizes, int n_in,
                              void* d_out, int out_size, void* d_ws, size_t ws_size,
                              hipStream_t stream) {
  const float* z  = (const float*)d_in[0];
  const int*   ei = (const int*)d_in[1];
  const float* W1 = (const float*)d_in[2];
  const float* b1 = (const float*)d_in[3];
  const float* W2 = (const float*)d_in[4];
  const float* b2 = (const float*)d_in[5];
  float* out = (float*)d_out;

  int Nn = in_sizes[0] / FIN;  // 100000
  int E  = in_sizes[1] / 2;    // 3200000
  const int* src = ei;
  const int* dst = ei + E;

  // workspace layout: bufA | bufB | deg(int) | dinv(float)
  char* ws = (char*)d_ws;
  size_t nodeCh = (size_t)Nn * CH * sizeof(float);
  float* bufA = (float*)ws;
  float* bufB = (float*)(ws + nodeCh);
  int*   degi = (int*)(ws + 2 * nodeCh);
  float* dinv = (float*)(ws + 2 * nodeCh + (size_t)Nn * sizeof(int));

  int total4 = Nn * (CH / 4);
  int mtiles = (Nn + 15) / 16;
  int eblk = (E + 7) / 8;

  // normalization coefficients
  deg_init_kernel<<<(Nn + 255) / 256, 256, 0, stream>>>(degi, Nn);
  deg_count_kernel<<<(E + 255) / 256, 256, 0, stream>>>(dst, degi, E);
  dinv_kernel<<<(Nn + 255) / 256, 256, 0, stream>>>(degi, dinv, Nn);

  // ---- layer 1 ----
  gemm16_kernel<<<mtiles, 256, 0, stream>>>(z, W1, bufA, Nn);  // h1 = z@W1
  self_bias_kernel<<<(total4 + 255) / 256, 256, 0, stream>>>(
      (const float4*)bufA, dinv, (const float4*)b1, (float4*)bufB, total4);
  scatter_kernel<<<eblk, 256, 0, stream>>>(bufA, src, dst, dinv, bufB, E);  // agg1 in bufB

  // ---- layer 2 ----
  gemm256_kernel<<<mtiles, 256, 0, stream>>>(bufB, W2, bufA, Nn);  // h2 = agg1@W2
  self_bias_kernel<<<(total4 + 255) / 256, 256, 0, stream>>>(
      (const float4*)bufA, dinv, (const float4*)b2, (float4*)bufB, total4);
  scatter_kernel<<<eblk, 256, 0, stream>>>(bufA, src, dst, dinv, bufB, E);  // agg2 in bufB

  // ---- decoder ----
  edge_dot_kernel<<<eblk, 256, 0, stream>>>(bufB, src, dst, out, E);
}